// PoolEncoder_29910152249928
// MI455X (gfx1250) — compile-verified
//
#include <hip/hip_runtime.h>
#include <math.h>

typedef __attribute__((ext_vector_type(2))) float v2f;
typedef __attribute__((ext_vector_type(8))) float v8f;
typedef __attribute__((ext_vector_type(4))) unsigned int u32x4;
typedef __attribute__((ext_vector_type(4))) int i32x4;
typedef __attribute__((ext_vector_type(8))) int i32x8;

#define D 128  // feature dim (validated on host)

// ---------------- utility kernels ----------------

__global__ void zero_f32(float* __restrict__ p, long long n) {
    long long i = (long long)blockIdx.x * blockDim.x + threadIdx.x;
    if (i < n) p[i] = 0.0f;
}

__global__ void init_i32(int* __restrict__ p, int v, int n) {
    int i = blockIdx.x * blockDim.x + threadIdx.x;
    if (i < n) p[i] = v;
}

// ---------------- TDM: 1-D contiguous tile (nelem 4-byte words) -> LDS ----------------
// D# layout per CDNA5 ISA ch.8: group0 = {count, lds_addr, global_addr[56:0], type=2},
// group1 = {data_size=4B, tensor_dim0=tile_dim0=nelem (1-D), trivial strides}.
// Groups 2/3 zero (tensor rank <= 2). Tracked on TENSORcnt.
__device__ __forceinline__ void tdm_load_1d(unsigned lds_off, const void* gptr, unsigned nelem) {
    unsigned long long ga = (unsigned long long)(size_t)gptr;
    u32x4 g0;
    g0[0] = 1u;                                            // count=1, user descriptor
    g0[1] = lds_off;                                       // lds_addr (bytes)
    g0[2] = (unsigned)(ga & 0xFFFFFFFFu);                  // global_addr[31:0]
    g0[3] = (unsigned)((ga >> 32) & 0x01FFFFFFu) | (2u << 30);  // addr[56:32] | type=2
    i32x8 g1;
    g1[0] = (int)(2u << 16);                               // data_size code 2 = 4 bytes
    g1[1] = (int)((nelem & 0xFFFFu) << 16);                // tensor_dim0[15:0]
    g1[2] = (int)((1u << 16) | ((nelem >> 16) & 0xFFFFu)); // tensor_dim1=1 | dim0[31:16]
    g1[3] = (int)((nelem & 0xFFFFu) << 16);                // tile_dim0 = nelem
    g1[4] = 0;                                             // tile_dim1/2 unused
    g1[5] = (int)nelem;                                    // tensor_dim0_stride
    g1[6] = 0;
    g1[7] = 0;
    i32x4 z4 = {0, 0, 0, 0};
#if __clang_major__ >= 23
    i32x8 z8 = {0, 0, 0, 0, 0, 0, 0, 0};
    __builtin_amdgcn_tensor_load_to_lds(g0, g1, z4, z4, z8, 0);
#else
    __builtin_amdgcn_tensor_load_to_lds(g0, g1, z4, z4, 0);
#endif
}

// ---------------- stage 1: embedding mean-pool ----------------
__global__ void embed_meanpool(const int* __restrict__ x, const float* __restrict__ embed,
                               float* __restrict__ h, int n_nodes, int L) {
    int node = blockIdx.x;
    if (node >= n_nodes) return;
    int d = threadIdx.x;
    const int* toks = x + (long long)node * L;
    float s = 0.0f;
    int cnt = 0;
    for (int l = 0; l < L; ++l) {
        int t = toks[l];
        if (t != 0) { s += embed[(long long)t * D + d]; ++cnt; }
    }
    float inv = 1.0f / (float)(cnt > 0 ? cnt : 1);
    h[(long long)node * D + d] = s * inv;
}

// ---------------- degree count ----------------
__global__ void deg_count(const int* __restrict__ dst, float* __restrict__ deg, int E) {
    int t = blockIdx.x * blockDim.x + threadIdx.x;
    if (t < E) atomicAdd(&deg[dst[t]], 1.0f);
}

// ---------------- edge scatter: agg[dst] += h[src] ----------------
__global__ void edge_scatter(const int* __restrict__ src, const int* __restrict__ dst,
                             const float* __restrict__ h, float* __restrict__ agg,
                             long long total /* E * D */) {
    long long t = (long long)blockIdx.x * blockDim.x + threadIdx.x;
    if (t >= total) return;
    int e = (int)(t >> 7);     // D == 128
    int d = (int)(t & 127);
    atomicAdd(&agg[(long long)dst[e] * D + d], h[(long long)src[e] * D + d]);
}

// ---------------- agg /= max(deg,1) ----------------
__global__ void normalize_agg(float* __restrict__ agg, const float* __restrict__ deg, long long total) {
    long long t = (long long)blockIdx.x * blockDim.x + threadIdx.x;
    if (t >= total) return;
    agg[t] = agg[t] / fmaxf(deg[t >> 7], 1.0f);
}

// ---------------- SAGE layer GEMM: hout = relu(aggN@Wl + b + h@Wr) ----------------
// 256 threads = 8 wave32s; block b owns rows [16b,16b+16); wave w owns cols [16w,16w+16).
// Wl and Wr (64 KB each) are staged into LDS once per block by the Tensor Data Mover;
// A fragments stream from global, B fragments come from LDS (ds_load), accumulation is
// full-precision fp32 WMMA (V_WMMA_F32_16X16X4_F32), two matmuls fused per k-step.
__global__ void sage_gemm_relu(const float* __restrict__ aggN, const float* __restrict__ h,
                               const float* __restrict__ Wl, const float* __restrict__ Wr,
                               const float* __restrict__ bias, float* __restrict__ hout,
                               int n_rows) {
    extern __shared__ float smem[];              // [2][128*128] floats = 128 KB
    float* WlL = smem;
    float* WrL = smem + D * D;

    if (threadIdx.x < 32) {                      // wave 0 drives the TDM (EXEC ignored)
        unsigned lds_base = (unsigned)(size_t)(void*)smem;  // flat LDS addr low 32b = offset
        tdm_load_1d(lds_base, Wl, D * D);
        tdm_load_1d(lds_base + D * D * sizeof(float), Wr, D * D);
        __builtin_amdgcn_s_wait_tensorcnt(0);    // TENSORcnt is per-wave
    }
    __syncthreads();                             // publish LDS to all 8 waves

    const int wave = threadIdx.x >> 5;           // 0..7 -> column tile
    const int lane = threadIdx.x & 31;
    const int m0 = blockIdx.x * 16;
    const int n0 = wave * 16;
    const int row = m0 + (lane & 15);            // A-fragment row for this lane
    const int col = n0 + (lane & 15);            // B/C/D column for this lane
    const int khalf = (lane >> 4) * 2;           // lanes 0-15: K+{0,1}; 16-31: K+{2,3}

    const int rclamp = row < n_rows ? row : (n_rows - 1);
    const float* __restrict__ arow = aggN + (long long)rclamp * D;
    const float* __restrict__ hrow = h + (long long)rclamp * D;

    v8f acc = {};
    for (int k0 = 0; k0 < D; k0 += 4) {
        v2f a, b;
        // neighbor-aggregate term: aggN @ Wl
        a.x = arow[k0 + khalf];
        a.y = arow[k0 + khalf + 1];
        b.x = WlL[(k0 + khalf) * D + col];
        b.y = WlL[(k0 + khalf + 1) * D + col];
        acc = __builtin_amdgcn_wmma_f32_16x16x4_f32(false, a, false, b, (short)0, acc,
                                                    false, false);
        // self term: h @ Wr
        a.x = hrow[k0 + khalf];
        a.y = hrow[k0 + khalf + 1];
        b.x = WrL[(k0 + khalf) * D + col];
        b.y = WrL[(k0 + khalf + 1) * D + col];
        acc = __builtin_amdgcn_wmma_f32_16x16x4_f32(false, a, false, b, (short)0, acc,
                                                    false, false);
    }

    const float bn = bias[col];
    // C/D layout: VGPR i -> row m0 + (lane<16 ? 0 : 8) + i, column = col
    const int rbase = m0 + ((lane >> 4) * 8);
#pragma unroll
    for (int i = 0; i < 8; ++i) {
        int r = rbase + i;
        if (r < n_rows) hout[(long long)r * D + col] = fmaxf(acc[i] + bn, 0.0f);
    }
}

// ---------------- supernode mean pool ----------------
__global__ void pool_scatter(const float* __restrict__ h, const int* __restrict__ pool_idx,
                             float* __restrict__ xp, float* __restrict__ cnt, int n_nodes) {
    long long t = (long long)blockIdx.x * blockDim.x + threadIdx.x;
    long long total = (long long)n_nodes * D;
    if (t >= total) return;
    int node = (int)(t >> 7);
    int d = (int)(t & 127);
    int p = pool_idx[node];
    atomicAdd(&xp[(long long)p * D + d], h[t]);
    if (d == 0) atomicAdd(&cnt[p], 1.0f);
}

__global__ void pool_div(float* __restrict__ xp, const float* __restrict__ cnt, int P) {
    long long t = (long long)blockIdx.x * blockDim.x + threadIdx.x;
    if (t >= (long long)P * D) return;
    xp[t] = xp[t] / fmaxf(cnt[t >> 7], 1.0f);
}

// ---------------- pooled adjacency (dense, no self loops) ----------------
__global__ void adj_fill(const int* __restrict__ src, const int* __restrict__ dst,
                         const int* __restrict__ pool_idx, float* __restrict__ adj,
                         int E, int P) {
    int t = blockIdx.x * blockDim.x + threadIdx.x;
    if (t >= E) return;
    int sp = pool_idx[src[t]];
    int dp = pool_idx[dst[t]];
    if (sp != dp) adj[(long long)sp * P + dp] = 1.0f;  // idempotent; races benign
}

// ---------------- pooled batch id via segment max ----------------
__global__ void batch_max(const int* __restrict__ node_batch, const int* __restrict__ pool_idx,
                          int* __restrict__ bmax, int n_nodes) {
    int t = blockIdx.x * blockDim.x + threadIdx.x;
    if (t < n_nodes) atomicMax(&bmax[pool_idx[t]], node_batch[t]);
}

__global__ void batch_to_f32(const int* __restrict__ bmax, float* __restrict__ out, int P) {
    int t = blockIdx.x * blockDim.x + threadIdx.x;
    if (t < P) out[t] = (float)bmax[t];
}

// ---------------- host orchestration ----------------

static inline int cdiv_i(long long a, long long b) { return (int)((a + b - 1) / b); }

extern "C" void kernel_launch(void* const* d_in, const int* in_sizes, int n_in,
                              void* d_out, int out_size, void* d_ws, size_t ws_size,
                              hipStream_t stream) {
    // input order: x, edge_index, pool_idx, node_batch, num_pooled, embed, W_l, W_r, b
    const int*   x          = (const int*)d_in[0];
    const int*   edge_index = (const int*)d_in[1];
    const int*   pool_idx   = (const int*)d_in[2];
    const int*   node_batch = (const int*)d_in[3];
    const float* embed      = (const float*)d_in[5];
    const float* W_l        = (const float*)d_in[6];
    const float* W_r        = (const float*)d_in[7];
    const float* bvec       = (const float*)d_in[8];

    const int N  = in_sizes[2];             // pool_idx has one entry per node
    const int L  = in_sizes[0] / N;
    const int E  = in_sizes[1] / 2;
    const int Dd = in_sizes[6] / in_sizes[8];  // (NL*D*D)/(NL*D)
    const int NL = in_sizes[8] / Dd;
    if (Dd != D) return;                    // kernels are specialized for D=128
    // out_size = P*D + P*P + P  ->  P^2 + (D+1)P - out_size = 0
    const int P = (int)((-(double)(D + 1) + sqrt((double)(D + 1) * (D + 1) +
                                                 4.0 * (double)out_size)) * 0.5 + 0.5);

    const int* src = edge_index;
    const int* dst = edge_index + E;

    // workspace carve-up (floats)
    float* h0  = (float*)d_ws;                       // [N, D]
    float* h1  = h0 + (long long)N * D;              // [N, D]
    float* agg = h1 + (long long)N * D;              // [N, D]
    float* deg = agg + (long long)N * D;             // [N]
    float* cnt = deg + N;                            // [P]
    int*   bmx = (int*)(cnt + P);                    // [P]
    (void)ws_size;

    float* out_xp   = (float*)d_out;                 // [P, D]
    float* out_adj  = out_xp + (long long)P * D;     // [P, P]
    float* out_bat  = out_adj + (long long)P * P;    // [P]

    const long long ND = (long long)N * D;
    const long long ED = (long long)E * D;
    const size_t gemm_lds = 2ull * D * D * sizeof(float);  // 128 KB (WGP has 320 KB)

    // degrees
    zero_f32<<<cdiv_i(N, 256), 256, 0, stream>>>(deg, N);
    deg_count<<<cdiv_i(E, 256), 256, 0, stream>>>(dst, deg, E);

    // embedding mean-pool
    embed_meanpool<<<N, D, 0, stream>>>(x, embed, h0, N, L);

    // SAGE layers (ping-pong h0/h1)
    float* hcur = h0;
    float* hnxt = h1;
    for (int layer = 0; layer < NL; ++layer) {
        zero_f32<<<cdiv_i(ND, 256), 256, 0, stream>>>(agg, ND);
        edge_scatter<<<cdiv_i(ED, 256), 256, 0, stream>>>(src, dst, hcur, agg, ED);
        normalize_agg<<<cdiv_i(ND, 256), 256, 0, stream>>>(agg, deg, ND);
        sage_gemm_relu<<<cdiv_i(N, 16), 256, gemm_lds, stream>>>(
            agg, hcur, W_l + (long long)layer * D * D, W_r + (long long)layer * D * D,
            bvec + (long long)layer * D, hnxt, N);
        float* t = hcur; hcur = hnxt; hnxt = t;
    }

    // supernode mean pool -> d_out[0 : P*D]
    zero_f32<<<cdiv_i((long long)P * D, 256), 256, 0, stream>>>(out_xp, (long long)P * D);
    zero_f32<<<cdiv_i(P, 256), 256, 0, stream>>>(cnt, P);
    pool_scatter<<<cdiv_i(ND, 256), 256, 0, stream>>>(hcur, pool_idx, out_xp, cnt, N);
    pool_div<<<cdiv_i((long long)P * D, 256), 256, 0, stream>>>(out_xp, cnt, P);

    // dense pooled adjacency -> d_out[P*D : P*D + P*P]
    zero_f32<<<cdiv_i((long long)P * P, 256), 256, 0, stream>>>(out_adj, (long long)P * P);
    adj_fill<<<cdiv_i(E, 256), 256, 0, stream>>>(src, dst, pool_idx, out_adj, E, P);

    // pooled batch ids -> d_out tail
    init_i32<<<cdiv_i(P, 256), 256, 0, stream>>>(bmx, (int)0x80000000, P);
    batch_max<<<cdiv_i(N, 256), 256, 0, stream>>>(node_batch, pool_idx, bmx, N);
    batch_to_f32<<<cdiv_i(P, 256), 256, 0, stream>>>(bmx, out_bat, P);
}